// ColorMapGenerator_87247965651646
// MI455X (gfx1250) — compile-verified
//
#include <hip/hip_runtime.h>
#include <math.h>

// ColorMapGenerator for MI455X (gfx1250):
// per-pixel 24-bit LUT gather + affine + tanh, NCHW in / NCHW out.
// Memory-gather-bound kernel (~1.4 GB effective HBM traffic -> ~60us at
// 23.3 TB/s). NT hints keep the 402MB LUT resident in the 192MB L2 while
// the 201MB of pixel traffic streams through with TH_NT.

#define HW_CONST (512u * 512u)        // H*W = 262144 (power of two)

// Native clang vector type: required by __builtin_nontemporal_{load,store}
typedef float vfloat4 __attribute__((ext_vector_type(4)));

// 12-byte LUT row, 4-byte aligned (DWORD-aligned b96 loads are legal on gfx1250)
struct __attribute__((packed, aligned(4))) Row3 { float r, g, b; };

__device__ __forceinline__ float gfx1250_tanh(float x) {
#if defined(__gfx1250__) && __has_builtin(__builtin_amdgcn_tanhf)
  return __builtin_amdgcn_tanhf(x);
#elif defined(__gfx1250__)
  float y;
  // v_tanh_f32 is a TRANS32 op: hardware needs one independent op / V_NOP
  // before the result is consumed (compiler can't see through inline asm).
  asm("v_tanh_f32 %0, %1\n\tv_nop" : "=v"(y) : "v"(x));
  return y;
#else
  return tanhf(x);   // host-side pass only
#endif
}

__global__ __launch_bounds__(256)
void ColorMapGenerator_87247965651646_kernel(const float* __restrict__ img,
                                             const float* __restrict__ weight,
                                             const float* __restrict__ bias,
                                             float* __restrict__ out,
                                             unsigned nPix /* B*H*W, multiple of 4 */) {
  const unsigned q = blockIdx.x * blockDim.x + threadIdx.x;  // quad-of-pixels
  const unsigned i = q << 2;                                 // first pixel
  if (i >= nPix) return;

  // Unsigned power-of-two math: shift + mask, no signed-division fixups.
  const unsigned b  = i >> 18;               // i / (H*W)
  const unsigned hw = i & (HW_CONST - 1u);   // i % (H*W)
  const size_t base = (size_t)b * (size_t)(3u * HW_CONST) + (size_t)hw;

  // Streaming channel loads: 3x global_load_b128 with TH_NT (don't pollute L2).
  const vfloat4 c0v = __builtin_nontemporal_load((const vfloat4*)(img + base));
  const vfloat4 c1v = __builtin_nontemporal_load((const vfloat4*)(img + base + HW_CONST));
  const vfloat4 c2v = __builtin_nontemporal_load((const vfloat4*)(img + base + 2u * HW_CONST));

  const float c0[4] = {c0v.x, c0v.y, c0v.z, c0v.w};
  const float c1[4] = {c1v.x, c1v.y, c1v.z, c1v.w};
  const float c2[4] = {c2v.x, c2v.y, c2v.z, c2v.w};

  float y0[4], y1[4], y2[4];

#pragma unroll
  for (int j = 0; j < 4; ++j) {
    const float xr = c0[j];
    const float xg = c1[j];
    const float xb = c2[j];

    // rgb = (x + 1) * 127.5 in [0, 255); explicit _rn ops block FMA
    // contraction so the truncated index matches the reference rounding.
    const float rr = __fmul_rn(__fadd_rn(xr, 1.0f), 127.5f);
    const float gg = __fmul_rn(__fadd_rn(xg, 1.0f), 127.5f);
    const float bb = __fmul_rn(__fadd_rn(xb, 1.0f), 127.5f);

    const float fidx = __fadd_rn(
        __fadd_rn(__fmul_rn(rr, 65536.0f), __fmul_rn(gg, 256.0f)), bb);
    const unsigned idx = (unsigned)(int)fidx;  // truncate == astype(int32)

    // Regular-temporal gathers: these are what we want cached in L2.
    const Row3 w  = *(const Row3*)(weight + 3u * idx);
    const Row3 bi = *(const Row3*)(bias   + 3u * idx);

    y0[j] = gfx1250_tanh(w.r * xr + bi.r);
    y1[j] = gfx1250_tanh(w.g * xg + bi.g);
    y2[j] = gfx1250_tanh(w.b * xb + bi.b);
  }

  // Streaming stores: 3x global_store_b128 with TH_NT.
  const vfloat4 o0 = {y0[0], y0[1], y0[2], y0[3]};
  const vfloat4 o1 = {y1[0], y1[1], y1[2], y1[3]};
  const vfloat4 o2 = {y2[0], y2[1], y2[2], y2[3]};
  __builtin_nontemporal_store(o0, (vfloat4*)(out + base));
  __builtin_nontemporal_store(o1, (vfloat4*)(out + base + HW_CONST));
  __builtin_nontemporal_store(o2, (vfloat4*)(out + base + 2u * HW_CONST));
}

extern "C" void kernel_launch(void* const* d_in, const int* in_sizes, int n_in,
                              void* d_out, int out_size, void* d_ws, size_t ws_size,
                              hipStream_t stream) {
  const float* img    = (const float*)d_in[0];   // [32, 3, 512, 512] f32
  const float* weight = (const float*)d_in[1];   // [256^3, 3] f32
  const float* bias   = (const float*)d_in[2];   // [256^3, 3] f32
  float* out = (float*)d_out;                    // [32, 3, 512, 512] f32

  const unsigned nPix  = (unsigned)(in_sizes[0] / 3);  // B*H*W = 8,388,608
  const unsigned quads = nPix >> 2;                    // 4 pixels per thread
  const int threads = 256;                             // 8 wave32 per block
  const int blocks  = (int)((quads + threads - 1) / threads);

  ColorMapGenerator_87247965651646_kernel<<<blocks, threads, 0, stream>>>(
      img, weight, bias, out, nPix);
}